// LowRankSig_HigherOrder_54365696032803
// MI455X (gfx1250) — compile-verified
//
#include <hip/hip_runtime.h>
#include <hip/hip_bf16.h>

typedef __attribute__((ext_vector_type(2))) float v2f;
typedef __attribute__((ext_vector_type(8))) float v8f;

#define T_LEN 2048
#define F_X   63      // raw X features (time channel prepended -> 64)
#define NCOL  640     // LEN_TENSORS(10) * UNITS(64)
#define NB    32      // batch

// ---------------------------------------------------------------------------
// Kernel 1: M[bt, k*64+u] = sum_f Xa[bt,f] * W[f, k*64+u]
// Xa[.,0] = time channel, Xa[.,f] = X[., f-1].
// Block = 256 threads = 8 waves. All 8 waves share one 16x64 A tile staged in
// LDS (row stride 65 floats to avoid bank conflicts); each wave owns one
// 16-column N tile and keeps its B columns in registers, reused across 4
// consecutive M tiles. K=64 -> 16 chained V_WMMA_F32_16X16X4_F32.
// Result (167 MB) lands in the 192 MB L2 for the scan kernel to consume.
// ---------------------------------------------------------------------------
__global__ __launch_bounds__(256)
void lrsig_gemm(const float* __restrict__ X,
                const float* __restrict__ W,
                float* __restrict__ M)
{
    __shared__ float As[16 * 65];   // 16 time rows x 64 features, padded

    const int tid   = threadIdx.x;
    const int lane  = tid & 31;
    const int wave  = tid >> 5;                 // 0..7
    const int nt    = blockIdx.y * 8 + wave;    // 0..39 N tile
    const int c0    = nt * 16;                  // first output column
    const int n     = lane & 15;                // column / A row within tile
    const int khalf = lane >> 4;                // 0: K0,K1  1: K2,K3

    // Preload this wave's B fragment registers: B[k_local, n] for all 16 K steps.
    float breg[32];
    const float* Wc = W + c0 + n;
    #pragma unroll
    for (int s = 0; s < 16; ++s) {
        const int f0 = 4 * s + 2 * khalf;
        breg[2*s]     = Wc[(size_t)(f0)     * NCOL];
        breg[2*s + 1] = Wc[(size_t)(f0 + 1) * NCOL];
    }

    const int srow = tid >> 4;   // staging: row 0..15
    const int scol = tid & 15;   // staging: 4 features per thread

    #pragma unroll 1
    for (int i = 0; i < 4; ++i) {
        const int mt  = blockIdx.x * 4 + i;     // M tile index (0..4095)
        const int bt0 = mt * 16;                // first global row
        const int b   = bt0 >> 11;              // batch (T_LEN = 2048)
        const int t0  = bt0 & (T_LEN - 1);      // local time of first row

        // ---- stage Xa tile (16 x 64) into LDS ----
        {
            const int tg = t0 + srow;
            const float* xr = X + (size_t)(b * T_LEN + tg) * F_X;
            if (scol == 0)
                As[srow * 65] = fmaf((float)tg, 2.0f / (float)(T_LEN - 1), -1.0f);
            #pragma unroll
            for (int q = 0; q < 4; ++q) {
                const int j = scol + 16 * q;
                if (j < F_X) As[srow * 65 + 1 + j] = xr[j];
            }
        }
        __syncthreads();

        // ---- 16 x 16 tile, K = 64 via 16 WMMA f32 16x16x4 ----
        v8f acc = {};
        #pragma unroll
        for (int s = 0; s < 16; ++s) {
            const int f0 = 4 * s + 2 * khalf;
            v2f a;
            a.x = As[n * 65 + f0];
            a.y = As[n * 65 + f0 + 1];
            v2f bm;
            bm.x = breg[2*s];
            bm.y = breg[2*s + 1];
            acc = __builtin_amdgcn_wmma_f32_16x16x4_f32(
                      false, a, false, bm, (short)0, acc, false, false);
        }

        // ---- store D: VGPR j -> row (j + 8*khalf), column c0+n ----
        float* outp = M + (size_t)bt0 * NCOL + c0 + n;
        #pragma unroll
        for (int j = 0; j < 8; ++j)
            outp[(size_t)(j + 8 * khalf) * NCOL] = acc[j];
        __syncthreads();
    }
}

// ---------------------------------------------------------------------------
// Kernel 2: time-diff + sequential signature recursion per (b,u).
// One block per batch, thread = one unit. M rows come from L2 (167 MB fits in
// the 192 MB global L2). Double-buffered row loads + distance-8 prefetch hide
// L2 latency; per step ~35 dependent VALU ops.
// State: 6 exclusive prefix accumulators (read-before-update == ecumsum).
// ---------------------------------------------------------------------------
__global__ __launch_bounds__(64)
void lrsig_scan(const float* __restrict__ M, float* __restrict__ out)
{
    const int b = blockIdx.x;
    const int u = threadIdx.x;                      // 0..63
    const float* base = M + (size_t)b * T_LEN * NCOL + u;

    const float inv2 = 0.5f, inv3 = 1.0f / 3.0f, inv4 = 0.25f;

    float prev[10], cur[10];
    #pragma unroll
    for (int k = 0; k < 10; ++k) prev[k] = base[(size_t)k * 64];
    #pragma unroll
    for (int k = 0; k < 10; ++k) cur[k]  = base[(size_t)NCOL + (size_t)k * 64];

    float S1 = 0.f, S2 = 0.f, S3 = 0.f, S4 = 0.f, S5 = 0.f, S6 = 0.f;
    float acc = 0.f;

    for (int t = 1; t < T_LEN; ++t) {
        // prefetch 8 rows ahead (L2-resident; pulls lines toward the WGP)
        if (t + 8 < T_LEN) {
            const float* pf = base + (size_t)(t + 8) * NCOL;
            #pragma unroll
            for (int k = 0; k < 10; ++k)
                __builtin_prefetch(pf + (size_t)k * 64, 0, 0);
        }
        // load next row (independent of the scan dependency chain)
        float nxt[10];
        if (t + 1 < T_LEN) {
            const float* rn = base + (size_t)(t + 1) * NCOL;
            #pragma unroll
            for (int k = 0; k < 10; ++k) nxt[k] = rn[(size_t)k * 64];
        }

        // diffed M values at this step
        float d[10];
        #pragma unroll
        for (int k = 0; k < 10; ++k) d[k] = cur[k] - prev[k];

        // level 0
        acc += d[0];

        // m = 1:  R=[M1]; i=1: A=M1 -> R=[M2*ecum(M1), M2*M1/2]
        {
            const float p1 = S1; S1 += d[1];
            acc += d[2] * p1 + d[2] * d[1] * inv2;
        }
        // m = 2
        {
            const float p2 = S2; S2 += d[3];
            const float r0 = d[4] * p2;
            const float r1 = d[4] * d[3] * inv2;
            const float p3 = S3; S3 += r0 + r1;
            acc += d[5] * p3 + d[5] * r0 * inv2 + d[5] * r1 * inv3;
        }
        // m = 3
        {
            const float p4 = S4; S4 += d[6];
            const float r0 = d[7] * p4;
            const float r1 = d[7] * d[6] * inv2;
            const float p5 = S5; S5 += r0 + r1;
            const float q0 = d[8] * p5;
            const float q1 = d[8] * r0 * inv2;
            const float q2 = d[8] * r1 * inv3;
            const float p6 = S6; S6 += q0 + q1 + q2;
            acc += d[9] * p6 + d[9] * q0 * inv2 + d[9] * q1 * inv3 + d[9] * q2 * inv4;
        }

        #pragma unroll
        for (int k = 0; k < 10; ++k) { prev[k] = cur[k]; cur[k] = nxt[k]; }
    }

    out[b * 64 + u] = acc;
}

// ---------------------------------------------------------------------------
// launch: d_in[0] = X (32*2048*63 f32), d_in[1] = kernel (64*10*64 f32)
// d_out = (32,64) f32.  d_ws holds M: 32*2048*640 f32 = 167,772,160 bytes.
// ---------------------------------------------------------------------------
extern "C" void kernel_launch(void* const* d_in, const int* in_sizes, int n_in,
                              void* d_out, int out_size, void* d_ws, size_t ws_size,
                              hipStream_t stream)
{
    const float* X = (const float*)d_in[0];
    const float* W = (const float*)d_in[1];
    float* out = (float*)d_out;
    float* Mws = (float*)d_ws;   // requires >= 32*2048*640*4 bytes

    // GEMM: 4096 M tiles grouped by 4 per block (x), 5 groups of 8 N tiles (y)
    dim3 ggrid(1024, 5, 1);
    lrsig_gemm<<<ggrid, 256, 0, stream>>>(X, W, Mws);

    // Scan: one block per batch, one thread per unit
    lrsig_scan<<<NB, 64, 0, stream>>>(Mws, out);
}